// MGNO1DTime_17927193494232
// MI455X (gfx1250) — compile-verified
//
#include <hip/hip_runtime.h>
#include <math.h>

// ---------------------------------------------------------------------------
// MGNO-1D-time, MI455X (gfx1250) fused implementation.
//
// Graph structure collapses (faithful to the reference's batch-mixing
// reshape) to a closed-form per-node combine:
//   agg[b,m] = H[b,m]                                          (b in {0,1})
//   agg[b,m] = H[b,m]/(1+2*d_m) + (2*d_m/sqrt(1+2*d_m))*H[b-2,m](b in {2,3})
// with d_m = (i>0)+(i<NX-1)+(t>0)+(t<T_OUT-1), i=m%NX, t=m/NX.
// Each layer: fp32 WMMA GEMM (16x16x4) + combine + LayerNorm + residual ReLU.
// X tile is staged into LDS by the Tensor Data Mover (3D tile, hardware
// padding reproduces the 132-float anti-conflict stride).
// ---------------------------------------------------------------------------

#define NXc   256
#define TINc  16
#define TOUTc 128
#define NGc   (NXc * TOUTc)      // 32768 nodes per batch
#define Bc    4
#define Nc    (Bc * NGc)         // 131072 rows
#define Wc    128                // hidden width
#define LDA   132                // LDS row stride (floats)

typedef float v2f __attribute__((ext_vector_type(2)));
typedef float v4f __attribute__((ext_vector_type(4)));
typedef float v8f __attribute__((ext_vector_type(8)));
typedef unsigned int u32x4 __attribute__((ext_vector_type(4)));
typedef int i32x4 __attribute__((ext_vector_type(4)));
typedef int i32x8 __attribute__((ext_vector_type(8)));
typedef __attribute__((address_space(3))) char lds_char;

// ---------------------------------------------------------------------------
// Kernel A: bilinear resize (T: 16->128, half-pixel, renormalized edge taps),
// feature build [u, P(4), xg, tg], and the 7x128 input GEMM.
// ---------------------------------------------------------------------------
__global__ __launch_bounds__(256) void mgno_input_kernel(
    const float* __restrict__ u0, const float* __restrict__ P,
    const float* __restrict__ W_in, const float* __restrict__ b_in,
    float* __restrict__ x)
{
  const int tid  = threadIdx.x;
  const int wave = tid >> 5;
  const int lane = tid & 31;
  const int n = blockIdx.x * 8 + wave;      // global row: b*NG + m
  const int b = n / NGc;
  const int m = n % NGc;
  const int ix = m >> 7;                    // feature convention: m = ix*T_OUT + it
  const int it = m & 127;

  const float src = (it + 0.5f) * (float(TINc) / float(TOUTc)) - 0.5f;
  const float f0  = floorf(src);
  const int   i0  = (int)f0;
  const float w   = src - f0;
  const float* urow = u0 + (size_t)(b * NXc + ix) * TINc;
  float uv;
  if (i0 < 0)                uv = urow[0];
  else if (i0 >= TINc - 1)   uv = urow[TINc - 1];
  else                       uv = urow[i0] * (1.0f - w) + urow[i0 + 1] * w;

  float feat[7];
  feat[0] = uv;
  feat[1] = P[b * 4 + 0];
  feat[2] = P[b * 4 + 1];
  feat[3] = P[b * 4 + 2];
  feat[4] = P[b * 4 + 3];
  feat[5] = (float)ix * (1.0f / (float)(NXc - 1));
  feat[6] = (float)it * (1.0f / (float)(TOUTc - 1));

  const int c = lane * 4;
  v4f acc = *(const v4f*)(b_in + c);
  #pragma unroll
  for (int f = 0; f < 7; ++f) {
    v4f wr = *(const v4f*)(W_in + f * Wc + c);
    acc.x += feat[f] * wr.x;
    acc.y += feat[f] * wr.y;
    acc.z += feat[f] * wr.z;
    acc.w += feat[f] * wr.w;
  }
  *(v4f*)(x + (size_t)n * Wc + c) = acc;
}

// ---------------------------------------------------------------------------
// Fused GCN layer.
// Block: 256 threads (8 wave32). Tile: 32 nodes x 4 batches = 128 rows.
// ---------------------------------------------------------------------------
__global__ __launch_bounds__(256) void mgno_layer_kernel(
    float* __restrict__ x,
    const float* __restrict__ Wg, const float* __restrict__ bg,
    const float* __restrict__ gamma, const float* __restrict__ beta)
{
  extern __shared__ float smem[];
  float* sWt = smem;                  // [128][LDA]: sWt[n*LDA+k] = W[k][n]
  float* sX  = smem + Wc * LDA;       // [128][LDA]: row r = 32*b + (m-m0)
  float* sH  = smem + 2 * Wc * LDA;

  const int tid  = threadIdx.x;
  const int wave = tid >> 5;
  const int lane = tid & 31;
  const int m0   = blockIdx.x * 32;

  // --- wave 0: kick off TDM load of the X tile into sX ---------------------
  // 3D tile: dim0 = 128 cols (contig), dim1 = 32 nodes (stride 128 elem),
  // dim2 = 4 batches (stride NG*128 elem). data_size = 4B.
  // pad_enable: +4 DWORDs after every 128 DWORDs -> LDS row stride 132.
  if (tid < 32) {
    const unsigned lds_addr =
        (unsigned)(unsigned long long)(lds_char*)sX;          // raw LDS bytes
    const unsigned long long ga =
        (unsigned long long)(uintptr_t)(x + (size_t)m0 * Wc); // tile start

    u32x4 g0;
    g0.x = 1u;                                  // count=1 (valid), user mode
    g0.y = lds_addr;                            // lds_addr [63:32]
    g0.z = (unsigned)(ga & 0xFFFFFFFFull);      // global_addr lo
    g0.w = (unsigned)((ga >> 32) & 0x1FFFFFFull)// global_addr [56:32]
         | (2u << 30);                          // type = 2 ("image")

    i32x8 g1;
    g1[0] = (2 << 16)                           // data_size = 4 bytes
          | (1 << 20)                           // pad_enable
          | (6 << 22)                           // pad_interval: 128 DWORDs
          | (3 << 25);                          // pad_amount: 4 DWORDs
    g1[1] = (int)(128u << 16);                  // tensor_dim0 = 128 (lo16)
    g1[2] = (int)(32u << 16);                   // tensor_dim1 = 32  (lo16)
    g1[3] = (int)(128u << 16);                  // tile_dim0 = 128
    g1[4] = 32 | (4 << 16);                     // tile_dim1=32, tile_dim2=4
    g1[5] = Wc;                                 // tensor_dim0_stride = 128
    g1[6] = 0;                                  // stride0 hi | stride1 lo16
    g1[7] = (int)((unsigned)(NGc) * Wc >> 16);  // stride1[47:16] = 0x40

    i32x4 g2;
    g2[0] = Bc;                                 // tensor_dim2 = 4
    g2[1] = 0; g2[2] = 0; g2[3] = 0;            // no dim3 / no iterate
    i32x4 g3 = {0, 0, 0, 0};
    i32x8 g4 = {0, 0, 0, 0, 0, 0, 0, 0};        // unused trailing group

    __builtin_amdgcn_tensor_load_to_lds(g0, g1, g2, g3, g4, 0);
  }

  // --- all waves: stage W transposed (concurrent with the TDM DMA) ---------
  {
    const int k     = tid >> 1;
    const int nbase = (tid & 1) * 64;
    const float* wrow = Wg + (size_t)k * Wc;
    #pragma unroll
    for (int j = 0; j < 64; j += 4) {
      v4f wv = *(const v4f*)(wrow + nbase + j);
      sWt[(nbase + j + 0) * LDA + k] = wv.x;
      sWt[(nbase + j + 1) * LDA + k] = wv.y;
      sWt[(nbase + j + 2) * LDA + k] = wv.z;
      sWt[(nbase + j + 3) * LDA + k] = wv.w;
    }
  }
  if (tid < 32) __builtin_amdgcn_s_wait_tensorcnt(0);
  __syncthreads();   // DMA complete (wave 0 arrived) + sWt visible

  // --- GEMM: wave owns a 16-row strip; A fragments hoisted across tiles ----
  {
    const int mrow  = 16 * wave + (lane & 15);
    const int khalf = (lane >> 4) * 2;          // fragment K sub-offset
    const int rbase = 16 * wave + ((lane >> 4) ? 8 : 0);

    v2f afr[32];
    #pragma unroll
    for (int k4 = 0; k4 < 32; ++k4)
      afr[k4] = *(const v2f*)(sX + mrow * LDA + 4 * k4 + khalf);

    for (int ct = 0; ct < 8; ++ct) {
      const int ncol = 16 * ct + (lane & 15);
      v8f acc = {0.f, 0.f, 0.f, 0.f, 0.f, 0.f, 0.f, 0.f};
      #pragma unroll
      for (int k4 = 0; k4 < 32; ++k4) {
        v2f bf = *(const v2f*)(sWt + ncol * LDA + 4 * k4 + khalf);
        acc = __builtin_amdgcn_wmma_f32_16x16x4_f32(
            false, afr[k4], false, bf, (short)0, acc, false, false);
      }
      const int cout = 16 * ct + (lane & 15);
      #pragma unroll
      for (int rr = 0; rr < 8; ++rr)
        sH[(rbase + rr) * LDA + cout] = acc[rr];
    }
  }
  __syncthreads();

  // --- combine + bias + LayerNorm + residual ReLU (2 threads per row) ------
  {
    const int r     = tid >> 1;
    const int cbase = (tid & 1) * 64;
    const int b     = r >> 5;
    const int mm    = m0 + (r & 31);

    float c0 = 1.0f, c1 = 0.0f;
    if (b >= 2) {
      const int gi = mm % NXc;             // edge convention: m = t*NX + i
      const int gt = mm / NXc;
      const int d  = (gi > 0) + (gi < NXc - 1) + (gt > 0) + (gt < TOUTc - 1);
      const float deg = 1.0f + 2.0f * (float)d;
      c0 = 1.0f / deg;
      c1 = (2.0f * (float)d) * rsqrtf(deg);
    }
    const float* hr = sH + r * LDA;
    const float* hs = sH + ((b >= 2) ? (r - 64) : r) * LDA;  // c1==0 if b<2

    float s1 = 0.f, s2 = 0.f;
    #pragma unroll
    for (int j = 0; j < 64; ++j) {
      const int c = cbase + j;
      const float v = c0 * hr[c] + c1 * hs[c] + bg[c];
      s1 += v;
      s2 += v * v;
    }
    s1 += __shfl_xor(s1, 1, 32);
    s2 += __shfl_xor(s2, 1, 32);
    const float mean = s1 * (1.0f / (float)Wc);
    const float var  = s2 * (1.0f / (float)Wc) - mean * mean;   // biased var
    const float rstd = rsqrtf(var + 1e-5f);

    float* outrow = x + ((size_t)b * NGc + mm) * Wc;
    #pragma unroll
    for (int j = 0; j < 64; ++j) {
      const int c = cbase + j;
      const float v = c0 * hr[c] + c1 * hs[c] + bg[c];
      const float y = (v - mean) * rstd * gamma[c] + beta[c];
      const float o = y + sX[r * LDA + c];
      outrow[c] = fmaxf(o, 0.0f);
    }
  }
}

// ---------------------------------------------------------------------------
// Output projection: out[n] = x[n,:] . W_out + b_out   (128 -> 1)
// ---------------------------------------------------------------------------
__global__ __launch_bounds__(256) void mgno_output_kernel(
    const float* __restrict__ x, const float* __restrict__ W_out,
    const float* __restrict__ b_out, float* __restrict__ out)
{
  const int n = blockIdx.x * 256 + threadIdx.x;
  const float* row = x + (size_t)n * Wc;
  float acc = b_out[0];
  #pragma unroll
  for (int c = 0; c < Wc; c += 4) {
    v4f xv = *(const v4f*)(row + c);
    v4f wv = *(const v4f*)(W_out + c);
    acc += xv.x * wv.x + xv.y * wv.y + xv.z * wv.z + xv.w * wv.w;
  }
  out[n] = acc;
}

// ---------------------------------------------------------------------------
extern "C" void kernel_launch(void* const* d_in, const int* in_sizes, int n_in,
                              void* d_out, int out_size, void* d_ws, size_t ws_size,
                              hipStream_t stream)
{
  const float* u0    = (const float*)d_in[0];
  const float* P     = (const float*)d_in[1];
  const float* W_in  = (const float*)d_in[2];
  const float* b_in  = (const float*)d_in[3];
  const float* Wg    = (const float*)d_in[4];   // (4,128,128)
  const float* bg    = (const float*)d_in[5];   // (4,128)
  const float* gamma = (const float*)d_in[6];   // (4,128)
  const float* beta  = (const float*)d_in[7];   // (4,128)
  const float* W_out = (const float*)d_in[8];   // (128,1)
  const float* b_out = (const float*)d_in[9];

  float* x = (float*)d_ws;                      // N x 128 fp32 = 64 MB scratch

  const size_t lds_bytes = (size_t)3 * Wc * LDA * sizeof(float);  // ~203 KB
  (void)hipFuncSetAttribute((const void*)mgno_layer_kernel,
                            hipFuncAttributeMaxDynamicSharedMemorySize,
                            (int)lds_bytes);

  mgno_input_kernel<<<Nc / 8, 256, 0, stream>>>(u0, P, W_in, b_in, x);
  for (int l = 0; l < 4; ++l) {
    mgno_layer_kernel<<<NGc / 32, 256, lds_bytes, stream>>>(
        x, Wg + (size_t)l * Wc * Wc, bg + l * Wc, gamma + l * Wc, beta + l * Wc);
  }
  mgno_output_kernel<<<Nc / 256, 256, 0, stream>>>(x, W_out, b_out, (float*)d_out);
}